// W2LossLegacy_24842090840387
// MI455X (gfx1250) — compile-verified
//
#include <hip/hip_runtime.h>

// Problem constants (from setup_inputs): N=2048 traces, T=4096 samples, P=256 knots.
#define T_LEN 4096
#define P_KN  256
#define EPS_F 1e-8f
#define BLK   256
#define EPT   (T_LEN / BLK)   // 16 elements per thread

typedef __attribute__((ext_vector_type(2))) float        v2f;
typedef __attribute__((ext_vector_type(4))) float        v4f;
typedef __attribute__((ext_vector_type(8))) float        v8f;
typedef __attribute__((ext_vector_type(4))) unsigned int v4u;
typedef __attribute__((ext_vector_type(8))) int          v8i;
typedef __attribute__((ext_vector_type(4))) int          v4i;

// ---------------------------------------------------------------------------
// TDM: DMA one row of 256 contiguous f32 (1 KB) from global to LDS.
// D# per CDNA5 ISA §8.3/8.4:
//   group0: count=1 | lds_addr[63:32] | global_addr[120:64] | type=2[127:126]
//   group1: data_size=4B | tensor_dim0=256, tensor_dim1=1 | tile 256x1 |
//           tensor_dim0_stride=256
//   groups 2/3 (+extra group on this toolchain): zero (tensor rank <= 2)
// Issue from ONE wave only (TDM ignores EXEC; one issue per executing wave).
// Completion tracked with TENSORcnt.
// ---------------------------------------------------------------------------
__device__ __forceinline__ void tdm_load_row256(const float* gsrc,
                                                unsigned int lds_addr) {
  const unsigned long long ga = (unsigned long long)(uintptr_t)gsrc;
  v4u g0;
  g0.x = 1u;                                              // count=1 (valid user D#)
  g0.y = lds_addr;                                        // LDS byte address
  g0.z = (unsigned int)ga;                                // global_addr[31:0]
  g0.w = (unsigned int)((ga >> 32) & 0x1FFFFFFull)        // global_addr[56:32]
         | (2u << 30);                                    // type = 2 ("image")
  v8i g1;
  g1[0] = 0x20000;            // workgroup_mask=0, data_size=2 (4 bytes)
  g1[1] = (int)(256u << 16);  // tensor_dim0[15:0]  -> bits 63:48   (=256)
  g1[2] = (int)(1u   << 16);  // tensor_dim0 hi = 0 | tensor_dim1[15:0]=1
  g1[3] = (int)(256u << 16);  // tensor_dim1 hi = 0 | tile_dim0 = 256
  g1[4] = 1;                  // tile_dim1 = 1, tile_dim2 = 0
  g1[5] = 256;                // tensor_dim0_stride low 32 (=256 elements)
  g1[6] = 0;                  // stride hi / tensor_dim1_stride lo
  g1[7] = 0;
  const v4i z4 = {0, 0, 0, 0};
  const v8i z8 = {0, 0, 0, 0, 0, 0, 0, 0};
  // amdgpu-toolchain (clang-23) 6-arg form.
  __builtin_amdgcn_tensor_load_to_lds(g0, g1, z4, z4, z8, 0);
}

// ---------------------------------------------------------------------------
// Block-wide inclusive scan of sh_partial[256] -> sh_scan[256].
// Core: W = S * U via 4x chained v_wmma_f32_16x16x4_f32 (K accumulated 4 at a
// time), where S[r][c] = partial[16r+c] and U is upper-triangular ones.
// Every wave computes redundantly so EXEC is all-ones (WMMA requirement);
// only wave 0 writes the result.
// ---------------------------------------------------------------------------
__device__ __forceinline__ void block_scan256(const float* sh_partial,
                                              float* sh_scan,
                                              float* sh_rowsum,
                                              float* sh_rowoff) {
  const int tid  = threadIdx.x;
  const int lane = tid & 31;
  const int mcol = lane & 15;   // A-row / B-col / D-col index
  const int hi   = lane >> 4;   // 0: lanes 0-15, 1: lanes 16-31

  v8f acc = {};
#pragma unroll
  for (int k0 = 0; k0 < 16; k0 += 4) {
    const int kb = k0 + (hi << 1);          // this lane's K pair within chunk
    v2f a, b;
    // A = S (data): A[M][K], M = lane&15
    a.x = sh_partial[16 * mcol + kb];
    a.y = sh_partial[16 * mcol + kb + 1];
    // B = U (upper-tri incl. diagonal): U[k][n] = (k <= n)
    b.x = (kb     <= mcol) ? 1.0f : 0.0f;
    b.y = (kb + 1 <= mcol) ? 1.0f : 0.0f;
    acc = __builtin_amdgcn_wmma_f32_16x16x4_f32(false, a, false, b,
                                                (short)0, acc, false, false);
  }
  // acc[g] = W[row = g + 8*hi][col = mcol] : row-wise inclusive scan of S.
  if (mcol == 15) {             // lanes 15 & 31 hold full row totals
#pragma unroll
    for (int g = 0; g < 8; ++g) sh_rowsum[g + 8 * hi] = acc[g];
  }
  __syncthreads();
  if (tid == 0) {               // 16-entry exclusive scan of row totals
    float r = 0.0f;
#pragma unroll
    for (int g = 0; g < 16; ++g) { sh_rowoff[g] = r; r += sh_rowsum[g]; }
  }
  __syncthreads();
  if (tid < 32) {               // wave 0 writes final inclusive scan
#pragma unroll
    for (int g = 0; g < 8; ++g) {
      const int row = g + 8 * hi;
      sh_scan[16 * row + mcol] = acc[g] + sh_rowoff[row];
    }
  }
  __syncthreads();
}

// ---------------------------------------------------------------------------
// Kernel 0: zero the output accumulator.
// ---------------------------------------------------------------------------
__global__ void w2_init_kernel(float* out) { out[0] = 0.0f; }

// ---------------------------------------------------------------------------
// Kernel 1: per-row obs processing -> quantile knots q and spline moments M.
// cumtrap identity used: c_i = dx*(S_i - 0.5*(y_0 + y_i)), S = incl. prefix
// sum of y; dx cancels after normalization.
// ---------------------------------------------------------------------------
__global__ void __launch_bounds__(BLK)
obs_prep_kernel(const float* __restrict__ obs, const float* __restrict__ t,
                const float* __restrict__ p, float* __restrict__ qbuf,
                float* __restrict__ Mbuf) {
  __shared__ float sh_cdf[T_LEN];
  __shared__ float sh_partial[BLK];
  __shared__ float sh_scan[BLK];
  __shared__ float sh_rowsum[16];
  __shared__ float sh_rowoff[16];
  __shared__ float sh_q[P_KN];
  __shared__ float sh_cp[P_KN];
  __shared__ float sh_dp[P_KN];
  __shared__ float sh_M[P_KN];
  __shared__ float sh_y0, sh_yl;

  const int row  = blockIdx.x;
  const int tid  = threadIdx.x;
  const int base = tid * EPT;
  const float* src = obs + (size_t)row * T_LEN;

  __builtin_prefetch(src + base, 0, 1);       // global_prefetch_b8

  float w[EPT], s[EPT];
#pragma unroll
  for (int i = 0; i < EPT; i += 4) {
    // row data is read exactly once across the dispatch -> non-temporal
    const v4f v =
        __builtin_nontemporal_load((const v4f*)(src + base) + (i >> 2));
    w[i + 0] = v.x * v.x + EPS_F;
    w[i + 1] = v.y * v.y + EPS_F;
    w[i + 2] = v.z * v.z + EPS_F;
    w[i + 3] = v.w * v.w + EPS_F;
  }
  float run = 0.0f;
#pragma unroll
  for (int i = 0; i < EPT; ++i) { run += w[i]; s[i] = run; }
  sh_partial[tid] = run;
  if (tid == 0)       sh_y0 = w[0];
  if (tid == BLK - 1) sh_yl = w[EPT - 1];
  __syncthreads();

  block_scan256(sh_partial, sh_scan, sh_rowsum, sh_rowoff);

  const float excl = (tid == 0) ? 0.0f : sh_scan[tid - 1];
  const float y0   = sh_y0, yl = sh_yl;
  const float Ctot = sh_scan[BLK - 1] - 0.5f * (y0 + yl);
  const float invC = 1.0f / Ctot;
#pragma unroll
  for (int i = 0; i < EPT; ++i) {
    const float Si = excl + s[i];
    sh_cdf[base + i] = (Si - 0.5f * (y0 + w[i])) * invC;   // normalized CDF
  }
  __syncthreads();

  // searchsorted(cdf, p[tid], side='left'); clip [1, T-2]; q = t[idx]
  {
    const float pv = p[tid];
    int lo = 0, hiB = T_LEN;
    while (lo < hiB) {
      const int mid = (lo + hiB) >> 1;
      if (sh_cdf[mid] < pv) lo = mid + 1; else hiB = mid;
    }
    int idx = lo;
    idx = idx < 1 ? 1 : (idx > T_LEN - 2 ? T_LEN - 2 : idx);
    const float qv = t[idx];
    sh_q[tid] = qv;
    qbuf[(size_t)row * P_KN + tid] = qv;
  }
  __syncthreads();

  // Natural cubic spline: tridiag (h*(4I+E)) Thomas solve, thread 0 serial;
  // 2048 blocks run this in parallel so the latency is hidden.
  if (tid == 0) {
    const float h = p[1] - p[0];
    const int   n = P_KN - 2;
    float pc = 0.0f, pd = 0.0f;
    for (int i = 0; i < n; ++i) {
      const float rhs = 6.0f * (sh_q[i + 2] - 2.0f * sh_q[i + 1] + sh_q[i]) / h;
      const float den = (i == 0) ? (4.0f * h) : (4.0f * h - h * pc);
      const float m   = 1.0f / den;
      const float cp  = h * m;
      const float dp  = (i == 0) ? (rhs * m) : ((rhs - h * pd) * m);
      sh_cp[i] = cp; sh_dp[i] = dp;
      pc = cp; pd = dp;
    }
    float x = sh_dp[n - 1];
    sh_M[n] = x;                       // interior i -> M[i+1]
    for (int i = n - 2; i >= 0; --i) {
      x = sh_dp[i] - sh_cp[i] * x;
      sh_M[i + 1] = x;
    }
    sh_M[0] = 0.0f; sh_M[P_KN - 1] = 0.0f;   // natural BCs
  }
  __syncthreads();
  Mbuf[(size_t)row * P_KN + tid] = sh_M[tid];
}

// ---------------------------------------------------------------------------
// Kernel 2: per-row f pass -> transport eval + weighted reduction into out[0].
// q/M spline tables are staged global->LDS by the Tensor Data Mover,
// overlapped with the whole prefix-scan phase; one s_wait_tensorcnt before use.
// ---------------------------------------------------------------------------
__global__ void __launch_bounds__(BLK)
w2_loss_kernel(const float* __restrict__ f, const float* __restrict__ t,
               const float* __restrict__ p, const float* __restrict__ qbuf,
               const float* __restrict__ Mbuf, float* __restrict__ out) {
  __shared__ float sh_partial[BLK];
  __shared__ float sh_scan[BLK];
  __shared__ float sh_rowsum[16];
  __shared__ float sh_rowoff[16];
  __shared__ float sh_q[P_KN];
  __shared__ float sh_M[P_KN];
  __shared__ float sh_red[BLK / 32];
  __shared__ float sh_y0, sh_yl;

  const int row  = blockIdx.x;
  const int tid  = threadIdx.x;
  const int base = tid * EPT;
  const float* src = f + (size_t)row * T_LEN;

  // Kick off TDM loads of this row's spline tables (wave 0 issues once);
  // the DMA runs concurrently with the entire scan phase below.
  if (tid < 32) {
    tdm_load_row256(qbuf + (size_t)row * P_KN,
                    (unsigned int)(uintptr_t)&sh_q[0]);
    tdm_load_row256(Mbuf + (size_t)row * P_KN,
                    (unsigned int)(uintptr_t)&sh_M[0]);
  }
  __builtin_prefetch(src + base, 0, 1);       // global_prefetch_b8

  float w[EPT], s[EPT];
#pragma unroll
  for (int i = 0; i < EPT; i += 4) {
    const v4f v =
        __builtin_nontemporal_load((const v4f*)(src + base) + (i >> 2));
    w[i + 0] = v.x * v.x + EPS_F;
    w[i + 1] = v.y * v.y + EPS_F;
    w[i + 2] = v.z * v.z + EPS_F;
    w[i + 3] = v.w * v.w + EPS_F;
  }
  float run = 0.0f;
#pragma unroll
  for (int i = 0; i < EPT; ++i) { run += w[i]; s[i] = run; }
  sh_partial[tid] = run;
  if (tid == 0)       sh_y0 = w[0];
  if (tid == BLK - 1) sh_yl = w[EPT - 1];
  __syncthreads();

  block_scan256(sh_partial, sh_scan, sh_rowsum, sh_rowoff);

  // TDM completion: issuing wave drains TENSORcnt, barrier publishes LDS.
  if (tid < 32) __builtin_amdgcn_s_wait_tensorcnt(0);
  __syncthreads();

  const float excl = (tid == 0) ? 0.0f : sh_scan[tid - 1];
  const float y0   = sh_y0, yl = sh_yl;
  const float Ctot = sh_scan[BLK - 1] - 0.5f * (y0 + yl);
  const float invC = 1.0f / Ctot;

  const float dt        = t[1] - t[0];
  const float p0        = p[0];
  const float h         = p[1] - p[0];
  const float hinv      = 1.0f / h;
  const float pdf_scale = invC / dt;          // pdf_i = w_i * invC / dt
  const float sixth     = 1.0f / 6.0f;

  float lsum = 0.0f;
#pragma unroll
  for (int i = 0; i < EPT; i += 4) {
    const v4f tv = *((const v4f*)(t + base) + (i >> 2));
    const float tvals[4] = {tv.x, tv.y, tv.z, tv.w};
#pragma unroll
    for (int k = 0; k < 4; ++k) {
      const int   ii = i + k;
      const float Si = excl + s[ii];
      const float x  = (Si - 0.5f * (y0 + w[ii])) * invC;  // normalized CDF

      // interval on uniform p grid == searchsorted(p,x,'right')-1, clipped
      int j = (int)floorf((x - p0) * hinv);
      j = j < 0 ? 0 : (j > P_KN - 2 ? P_KN - 2 : j);

      const float qj  = sh_q[j],  qj1 = sh_q[j + 1];
      const float Mj  = sh_M[j],  Mj1 = sh_M[j + 1];
      const float dxq = x - (p0 + (float)j * h);
      const float b   = (qj1 - qj) * hinv - h * (2.0f * Mj + Mj1) * sixth;
      const float c   = 0.5f * Mj;
      const float d   = (Mj1 - Mj) * (hinv * sixth);
      const float tr  = qj + dxq * (b + dxq * (c + dxq * d));

      const float diff = tvals[k] - tr;
      const float g    = diff * diff * w[ii] * pdf_scale;
      const int   gi   = base + ii;
      const float wgt  = (gi == 0 || gi == T_LEN - 1) ? 0.5f : 1.0f;
      lsum += wgt * g;
    }
  }
  lsum *= dt;                                  // trapezoid dx factor

  // wave32 reduction, then cross-wave via LDS, then one atomic per block
#pragma unroll
  for (int off = 16; off > 0; off >>= 1) lsum += __shfl_down(lsum, off, 32);
  if ((tid & 31) == 0) sh_red[tid >> 5] = lsum;
  __syncthreads();
  if (tid == 0) {
    float tot = 0.0f;
#pragma unroll
    for (int wv = 0; wv < BLK / 32; ++wv) tot += sh_red[wv];
    atomicAdd(out, tot);                       // global_atomic_add_f32
  }
}

// ---------------------------------------------------------------------------
// Host entry
// ---------------------------------------------------------------------------
extern "C" void kernel_launch(void* const* d_in, const int* in_sizes, int n_in,
                              void* d_out, int out_size, void* d_ws, size_t ws_size,
                              hipStream_t stream) {
  const float* f   = (const float*)d_in[0];
  const float* obs = (const float*)d_in[1];
  const float* t   = (const float*)d_in[2];
  const float* p   = (const float*)d_in[3];
  float* out = (float*)d_out;

  const int N = in_sizes[0] / T_LEN;           // 2048

  float* qbuf = (float*)d_ws;                  // [N, P]
  float* Mbuf = qbuf + (size_t)N * P_KN;       // [N, P]

  w2_init_kernel<<<1, 1, 0, stream>>>(out);
  obs_prep_kernel<<<N, BLK, 0, stream>>>(obs, t, p, qbuf, Mbuf);
  w2_loss_kernel<<<N, BLK, 0, stream>>>(f, t, p, qbuf, Mbuf, out);
}